// GCN2_49478023250096
// MI455X (gfx1250) — compile-verified
//
#include <hip/hip_runtime.h>
#include <math.h>

#define NN 100000
#define NE 1600000
#define NG 1024

typedef __attribute__((ext_vector_type(2))) float v2f;
typedef __attribute__((ext_vector_type(8))) float v8f;

// ---------------- degree / norm ----------------
__global__ void k_init_deg(float* deg, int n) {
    int i = blockIdx.x * blockDim.x + threadIdx.x;
    if (i < n) deg[i] = 1.0f;                    // self-loop
}
__global__ void k_deg_accum(const int* __restrict__ dst, float* deg, int e) {
    int i = blockIdx.x * blockDim.x + threadIdx.x;
    if (i < e) atomicAdd(&deg[dst[i]], 1.0f);
}
__global__ void k_rsqrt_inplace(float* d, int n) {
    int i = blockIdx.x * blockDim.x + threadIdx.x;
    if (i < n) d[i] = rsqrtf(d[i]);
}
__global__ void k_norm(const int* __restrict__ src, const int* __restrict__ dst,
                       const float* __restrict__ dinv, float* __restrict__ nrm, int e) {
    int i = blockIdx.x * blockDim.x + threadIdx.x;
    if (i < e) nrm[i] = dinv[src[i]] * dinv[dst[i]];
}

// ---------------- pad first-layer operands to K=8 ----------------
__global__ void k_pad_x(const float* __restrict__ x, float* __restrict__ xp) {
    int t = blockIdx.x * blockDim.x + threadIdx.x;
    if (t < NN * 8) {
        int i = t >> 3, j = t & 7;
        xp[t] = (j < 5) ? x[i * 5 + j] : 0.0f;
    }
}
__global__ void k_pad_w1(const float* __restrict__ w, float* __restrict__ wp) {
    int t = blockIdx.x * blockDim.x + threadIdx.x;
    if (t < 8 * 16) {
        int k = t >> 4, c = t & 15;
        wp[t] = (k < 5) ? w[k * 16 + c] : 0.0f;
    }
}

// ---------------- WMMA f32 GEMM: C[M,N] = A[M,K] @ B[K,N] (+bias)(relu) -------
// One wave per 16x16 tile; K a compile-time multiple of 8 -> no bounds checks,
// fully unrolled chain of V_WMMA_F32_16X16X4_F32.
template <int K, int N, bool RELU_IN, bool RELU_OUT, bool HAS_BIAS>
__global__ void k_gemm_wmma(const float* __restrict__ A, const float* __restrict__ B,
                            float* __restrict__ C, const float* __restrict__ bias) {
    const int lane = threadIdx.x;        // 0..31, full wave, EXEC all-1s
    const int m0   = blockIdx.x * 16;
    const int n0   = blockIdx.y * 16;
    const int half = lane >> 4;          // lane-half selects K pair (ISA 7.12.2)
    const int idx  = lane & 15;
    const float* __restrict__ arow = A + (m0 + idx) * K;   // row-major A
    const float* __restrict__ bcol = B + n0 + idx;         // column of B
    v8f acc = {};
#pragma unroll
    for (int k0 = 0; k0 < K; k0 += 4) {
        const int ka = k0 + 2 * half;
        float2 av = *(const float2*)(arow + ka);           // 8B-aligned (K%8==0)
        v2f a, b;
        if constexpr (RELU_IN) {
            a.x = fmaxf(av.x, 0.0f);
            a.y = fmaxf(av.y, 0.0f);
        } else {
            a.x = av.x;
            a.y = av.y;
        }
        b.x = bcol[(ka + 0) * N];
        b.y = bcol[(ka + 1) * N];
        acc = __builtin_amdgcn_wmma_f32_16x16x4_f32(
                  false, a, false, b, (short)0, acc, false, false);
    }
    const int col = n0 + idx;
    float bv;
    if constexpr (HAS_BIAS) bv = bias[col];
#pragma unroll
    for (int r = 0; r < 8; ++r) {        // C/D: VGPR r -> rows r and r+8
        float v = acc[r];
        if constexpr (HAS_BIAS) v += bv;
        if constexpr (RELU_OUT) v = fmaxf(v, 0.0f);
        C[(m0 + r + 8 * half) * N + col] = v;
    }
}

// ---------------- self-loop + bias init of scatter output (float4) ----------
template <int D>
__global__ void k_selfloop_bias(const float* __restrict__ h, const float* __restrict__ dinv,
                                const float* __restrict__ bias, float* __restrict__ out) {
    constexpr int NC = D / 4;
    int t = blockIdx.x * blockDim.x + threadIdx.x;
    if (t < NN * NC) {
        int i  = t / NC;
        int c4 = t - i * NC;
        float di = dinv[i];
        float s  = di * di;
        float4 hv = ((const float4*)h)[t];
        float4 bv = ((const float4*)bias)[c4];
        float4 ov;
        ov.x = hv.x * s + bv.x;
        ov.y = hv.y * s + bv.y;
        ov.z = hv.z * s + bv.z;
        ov.w = hv.w * s + bv.w;
        ((float4*)out)[t] = ov;
    }
}

// ---------------- edge gather * norm -> atomic scatter-add ----------------
template <int D>
__global__ void k_scatter(const float* __restrict__ h, const int* __restrict__ src,
                          const int* __restrict__ dst, const float* __restrict__ nrm,
                          float* out) {
    constexpr int NC = D / 4;
    int t = blockIdx.x * blockDim.x + threadIdx.x;
    if (t >= NE * NC) return;
    int ed = t / NC;
    int c4 = t - ed * NC;
    float w = nrm[ed];
    float4 hv = ((const float4*)h)[src[ed] * NC + c4];
    float* o = out + dst[ed] * D + c4 * 4;
    atomicAdd(o + 0, hv.x * w);
    atomicAdd(o + 1, hv.y * w);
    atomicAdd(o + 2, hv.z * w);
    atomicAdd(o + 3, hv.w * w);
}

// ---------------- graph readout: relu + segment_max (bit-pattern atomicMax) --
__global__ void k_zero(float* p, int n) {
    int i = blockIdx.x * blockDim.x + threadIdx.x;
    if (i < n) p[i] = 0.0f;
}
__global__ void k_segmax(const float* __restrict__ x, const int* __restrict__ batch,
                         float* g) {
    int t = blockIdx.x * blockDim.x + threadIdx.x;
    if (t < NN * 128) {
        int i = t >> 7, c = t & 127;
        float v = fmaxf(x[t], 0.0f);     // relu fused; v >= 0 so int compare is valid
        atomicMax((int*)&g[batch[i] * 128 + c], __float_as_int(v));
    }
}

// ---------------- final 64->10 + softmax ----------------
__global__ void k_head(const float* __restrict__ h1, const float* __restrict__ lw2,
                       const float* __restrict__ lb2, float* __restrict__ out, int G) {
    int gi = blockIdx.x * blockDim.x + threadIdx.x;
    if (gi >= G) return;
    float l[10];
    float m = -1e30f;
#pragma unroll
    for (int j = 0; j < 10; ++j) {
        float acc = lb2[j];
#pragma unroll
        for (int k = 0; k < 64; ++k) acc += h1[gi * 64 + k] * lw2[k * 10 + j];
        l[j] = acc;
        m = fmaxf(m, acc);
    }
    float s = 0.0f;
#pragma unroll
    for (int j = 0; j < 10; ++j) { l[j] = __expf(l[j] - m); s += l[j]; }
    float inv = 1.0f / s;
#pragma unroll
    for (int j = 0; j < 10; ++j) out[gi * 10 + j] = l[j] * inv;
}

// ---------------- per-layer driver ----------------
template <int K, int D, bool RELU_IN>
static void run_layer(const float* in, const float* w, const float* b,
                      float* bufA, float* bufB, const float* dinv,
                      const int* src, const int* dst, const float* nrm,
                      hipStream_t stream) {
    dim3 gg(NN / 16, D / 16);
    k_gemm_wmma<K, D, RELU_IN, false, false><<<gg, 32, 0, stream>>>(in, w, bufA, nullptr);
    k_selfloop_bias<D><<<(NN * (D / 4) + 255) / 256, 256, 0, stream>>>(bufA, dinv, b, bufB);
    constexpr long total = (long)NE * (D / 4);
    k_scatter<D><<<(int)((total + 255) / 256), 256, 0, stream>>>(bufA, src, dst, nrm, bufB);
}

static inline size_t al256(size_t x) { return (x + 255) & ~(size_t)255; }

extern "C" void kernel_launch(void* const* d_in, const int* in_sizes, int n_in,
                              void* d_out, int out_size, void* d_ws, size_t ws_size,
                              hipStream_t stream) {
    (void)in_sizes; (void)n_in; (void)out_size; (void)ws_size;
    const float* x     = (const float*)d_in[0];
    const int*   ei    = (const int*)d_in[1];
    const int*   batch = (const int*)d_in[2];
    const float* W[6]; const float* Bb[6];
    for (int i = 0; i < 6; ++i) { W[i] = (const float*)d_in[3 + 2*i]; Bb[i] = (const float*)d_in[4 + 2*i]; }
    const float* lw1 = (const float*)d_in[15];
    const float* lb1 = (const float*)d_in[16];
    const float* lw2 = (const float*)d_in[17];
    const float* lb2 = (const float*)d_in[18];
    float* out = (float*)d_out;

    const int* src = ei;            // edge_index[0]
    const int* dst = ei + NE;       // edge_index[1]

    // workspace carve (~114 MB)
    char* p = (char*)d_ws;
    size_t off = 0;
    float* deg   = (float*)(p + off); off = al256(off + (size_t)NN * 4);          // becomes dinv
    float* nrm   = (float*)(p + off); off = al256(off + (size_t)NE * 4);
    float* bufA  = (float*)(p + off); off = al256(off + (size_t)NN * 128 * 4);    // h (post-GEMM)
    float* bufB  = (float*)(p + off); off = al256(off + (size_t)NN * 128 * 4);    // node features
    float* xpad  = (float*)(p + off); off = al256(off + (size_t)NN * 8 * 4);
    float* w1pad = (float*)(p + off); off = al256(off + (size_t)8 * 16 * 4);
    float* g     = (float*)(p + off); off = al256(off + (size_t)NG * 128 * 4);
    float* h1    = (float*)(p + off); off = al256(off + (size_t)NG * 64 * 4);

    // normalization: deg = 1 + indegree; dinv = rsqrt(deg); nrm[e] = dinv[s]*dinv[d]
    k_init_deg     <<<(NN + 255) / 256, 256, 0, stream>>>(deg, NN);
    k_deg_accum    <<<(NE + 255) / 256, 256, 0, stream>>>(dst, deg, NE);
    k_rsqrt_inplace<<<(NN + 255) / 256, 256, 0, stream>>>(deg, NN);
    const float* dinv = deg;
    k_norm         <<<(NE + 255) / 256, 256, 0, stream>>>(src, dst, dinv, nrm, NE);

    // pad layer-1 operands (K=5 -> K=8 with zeros)
    k_pad_x <<<(NN * 8 + 255) / 256, 256, 0, stream>>>(x, xpad);
    k_pad_w1<<<1, 128, 0, stream>>>(W[0], w1pad);

    // 6 GCN layers (dims padded: 8,16,32,48,64,96,128)
    run_layer<  8,  16, false>(xpad, w1pad, Bb[0], bufA, bufB, dinv, src, dst, nrm, stream);
    run_layer< 16,  32, true >(bufB, W[1],  Bb[1], bufA, bufB, dinv, src, dst, nrm, stream);
    run_layer< 32,  48, true >(bufB, W[2],  Bb[2], bufA, bufB, dinv, src, dst, nrm, stream);
    run_layer< 48,  64, true >(bufB, W[3],  Bb[3], bufA, bufB, dinv, src, dst, nrm, stream);
    run_layer< 64,  96, true >(bufB, W[4],  Bb[4], bufA, bufB, dinv, src, dst, nrm, stream);
    run_layer< 96, 128, true >(bufB, W[5],  Bb[5], bufA, bufB, dinv, src, dst, nrm, stream);

    // readout: relu + segment_max over graphs
    k_zero  <<<(NG * 128 + 255) / 256, 256, 0, stream>>>(g, NG * 128);
    k_segmax<<<(NN * 128 + 255) / 256, 256, 0, stream>>>(bufB, batch, g);

    // MLP head: 128->64 (WMMA, bias+relu), then 64->10 + softmax
    dim3 gm(NG / 16, 64 / 16);
    k_gemm_wmma<128, 64, false, true, true><<<gm, 32, 0, stream>>>(g, lw1, h1, lb1);
    k_head<<<(NG + 255) / 256, 256, 0, stream>>>(h1, lw2, lb2, out, NG);
}